// GATEncoder_87213605912769
// MI455X (gfx1250) — compile-verified
//
#include <hip/hip_runtime.h>
#include <math.h>

#define NEG_SLOPE 0.2f
#define BN_EPS 1e-5f
#define NUM_GRAPHS 16

typedef float v2f __attribute__((ext_vector_type(2)));
typedef float v8f __attribute__((ext_vector_type(8)));

// ---------------------------------------------------------------- utilities

__global__ void fill_kernel(float* __restrict__ p, float v, int n) {
  int i = blockIdx.x * blockDim.x + threadIdx.x;
  if (i < n) p[i] = v;
}

// float atomic max via monotonic int/uint trick (init must be -inf)
__device__ __forceinline__ void atomicMaxF(float* addr, float val) {
  if (val >= 0.0f) atomicMax((int*)addr, __float_as_int(val));
  else             atomicMin((unsigned int*)addr, __float_as_uint(val));
}

// ------------------------------------------------- WMMA fp32 GEMM (exact)
// C[M,N] = A[M,K] @ B[K,N], row-major. M%16==0, N%16==0, K%4==0.
// One wave computes one 16x16 tile with V_WMMA_F32_16X16X4_F32.
// Fragment layouts per CDNA5 ISA 7.12.2:
//   A 16x4 : lanes 0-15 -> M=lane, VGPR0=K+0, VGPR1=K+1 ; lanes 16-31 -> K+2,K+3
//   B 4x16 : lanes 0-15 -> N=lane, VGPR0=K+0, VGPR1=K+1 ; lanes 16-31 -> K+2,K+3
//   C/D    : VGPR r -> M=r (lanes 0-15) / M=8+r (lanes 16-31), N=lane%16
__global__ void wmma_gemm_f32(const float* __restrict__ A, const float* __restrict__ B,
                              float* __restrict__ C, int M, int N, int K) {
  int wave   = (blockIdx.x * blockDim.x + threadIdx.x) >> 5;
  int tilesN = N >> 4;
  int tm = wave / tilesN;
  int tn = wave - tm * tilesN;
  int row0 = tm << 4, col0 = tn << 4;
  if (row0 >= M) return;                 // wave-uniform guard (EXEC stays all-1)
  int lane = threadIdx.x & 31;
  int ll   = lane & 15;                  // lane within half
  int kh   = lane >> 4;                  // 0: K+0/K+1, 1: K+2/K+3

  v8f acc = {};
  const float* Arow = A + (size_t)(row0 + ll) * K;
  for (int k = 0; k < K; k += 4) {
    v2f a, b;
    a.x = Arow[k + 2 * kh + 0];
    a.y = Arow[k + 2 * kh + 1];
    b.x = B[(size_t)(k + 2 * kh + 0) * N + col0 + ll];
    b.y = B[(size_t)(k + 2 * kh + 1) * N + col0 + ll];
    acc = __builtin_amdgcn_wmma_f32_16x16x4_f32(false, a, false, b,
                                                (short)0, acc, false, false);
  }
#pragma unroll
  for (int r = 0; r < 8; ++r)
    C[(size_t)(row0 + r + 8 * kh) * N + col0 + ll] = acc[r];
}

// ---------------------------------------------- per-node attention scores
// ssrc[n,h] = sum_c h[n,h,c]*a_src[h,c] ; sdst likewise
__global__ void scores_kernel(const float* __restrict__ h,
                              const float* __restrict__ a_src,
                              const float* __restrict__ a_dst,
                              float* __restrict__ ssrc, float* __restrict__ sdst,
                              int N, int H, int C) {
  int i = blockIdx.x * blockDim.x + threadIdx.x;    // n*H + hh
  if (i >= N * H) return;
  int n = i / H, hh = i - n * H;
  const float* hp = h + (size_t)n * H * C + (size_t)hh * C;
  const float* as = a_src + (size_t)hh * C;
  const float* ad = a_dst + (size_t)hh * C;
  float s1 = 0.f, s2 = 0.f;
  for (int c = 0; c < C; ++c) { float v = hp[c]; s1 += v * as[c]; s2 += v * ad[c]; }
  ssrc[i] = s1; sdst[i] = s2;
}

// -------------------------------------- edge sweep 1: leaky-relu + seg-max
__global__ void edge_logits_kernel(const int* __restrict__ ei, int E, int Etot,
                                   const float* __restrict__ ssrc,
                                   const float* __restrict__ sdst,
                                   float* __restrict__ ebuf, float* __restrict__ m,
                                   int H) {
  int e = blockIdx.x * blockDim.x + threadIdx.x;
  if (e >= Etot) return;
  int s, d;
  if (e < E) { s = ei[e]; d = ei[E + e]; } else { s = d = e - E; }  // self loops
  for (int hh = 0; hh < H; ++hh) {
    float v = ssrc[s * H + hh] + sdst[d * H + hh];
    v = v > 0.f ? v : NEG_SLOPE * v;
    ebuf[(size_t)e * H + hh] = v;
    atomicMaxF(&m[d * H + hh], v);
  }
}

// ---------------------------------------- edge sweep 2: exp + seg-sum
__global__ void edge_exp_kernel(const int* __restrict__ ei, int E, int Etot,
                                float* __restrict__ ebuf, const float* __restrict__ m,
                                float* __restrict__ den, int H) {
  int e = blockIdx.x * blockDim.x + threadIdx.x;
  if (e >= Etot) return;
  int d = (e < E) ? ei[E + e] : e - E;
  for (int hh = 0; hh < H; ++hh) {
    float ex = expf(ebuf[(size_t)e * H + hh] - m[d * H + hh]);
    ebuf[(size_t)e * H + hh] = ex;
    atomicAdd(&den[d * H + hh], ex);
  }
}

// ------------------------- edge sweep 3: alpha-weighted message scatter
// one block per edge, blockDim = H*C; coalesced gather of h[src], atomic scatter
__global__ void edge_msg_kernel(const int* __restrict__ ei, int E,
                                const float* __restrict__ ex,
                                const float* __restrict__ den,
                                const float* __restrict__ hbuf,
                                float* __restrict__ agg, int H, int C) {
  int e = blockIdx.x;
  int c = threadIdx.x;                 // 0 .. H*C-1
  int hh = c / C;
  int s, d;
  if (e < E) { s = ei[e]; d = ei[E + e]; } else { s = d = e - E; }
  float alpha = ex[(size_t)e * H + hh] / (den[d * H + hh] + 1e-16f);
  atomicAdd(&agg[(size_t)d * H * C + c], hbuf[(size_t)s * H * C + c] * alpha);
}

// ------------------------------------- bias + batchnorm(eval) [+ ELU]
__global__ void post_kernel(float* __restrict__ x, const float* __restrict__ b,
                            const float* __restrict__ g, const float* __restrict__ be,
                            const float* __restrict__ rm, const float* __restrict__ rv,
                            int N, int C, int do_elu) {
  int i = blockIdx.x * blockDim.x + threadIdx.x;
  if (i >= N * C) return;
  int c = i % C;
  float v = x[i] + b[c];
  v = (v - rm[c]) * rsqrtf(rv[c] + BN_EPS) * g[c] + be[c];
  if (do_elu) v = v > 0.f ? v : expf(v) - 1.f;
  x[i] = v;
}

// ------------------------------------- per-graph mean pool (stage 1: sums)
__global__ void pool_accum_kernel(const float* __restrict__ o2,
                                  const int* __restrict__ batch,
                                  float* __restrict__ sums, int N, int C) {
  __shared__ float acc[NUM_GRAPHS * 128];
  for (int i = threadIdx.x; i < NUM_GRAPHS * C; i += blockDim.x) acc[i] = 0.f;
  __syncthreads();
  int n0 = blockIdx.x * 64;
  for (int idx = threadIdx.x; idx < 64 * C; idx += blockDim.x) {
    int i = idx / C, c = idx - i * C;
    int n = n0 + i;
    if (n < N) atomicAdd(&acc[batch[n] * C + c], o2[(size_t)n * C + c]);
  }
  __syncthreads();
  for (int i = threadIdx.x; i < NUM_GRAPHS * C; i += blockDim.x)
    atomicAdd(&sums[i], acc[i]);
}

__global__ void count_kernel(const int* __restrict__ batch, float* __restrict__ cnt, int N) {
  int n = blockIdx.x * blockDim.x + threadIdx.x;
  if (n < N) atomicAdd(&cnt[batch[n]], 1.0f);
}

__global__ void finalize_kernel(const float* __restrict__ sums,
                                const float* __restrict__ cnt,
                                float* __restrict__ out, int G, int C) {
  int i = blockIdx.x * blockDim.x + threadIdx.x;
  if (i < G * C) out[i] = sums[i] / fmaxf(cnt[i / C], 1.0f);
}

// ------------------------------------------------------------------ driver

static inline int cdiv(int a, int b) { return (a + b - 1) / b; }

extern "C" void kernel_launch(void* const* d_in, const int* in_sizes, int n_in,
                              void* d_out, int out_size, void* d_ws, size_t ws_size,
                              hipStream_t stream) {
  const float* x   = (const float*)d_in[0];
  const int*   ei  = (const int*)d_in[1];
  const int*   bat = (const int*)d_in[2];
  const float* W1  = (const float*)d_in[3];
  const float* as1 = (const float*)d_in[4];
  const float* ad1 = (const float*)d_in[5];
  const float* b1  = (const float*)d_in[6];
  const float* g1  = (const float*)d_in[7];
  const float* be1 = (const float*)d_in[8];
  const float* rm1 = (const float*)d_in[9];
  const float* rv1 = (const float*)d_in[10];
  const float* W2  = (const float*)d_in[11];
  const float* as2 = (const float*)d_in[12];
  const float* ad2 = (const float*)d_in[13];
  const float* b2  = (const float*)d_in[14];
  const float* g2  = (const float*)d_in[15];
  const float* be2 = (const float*)d_in[16];
  const float* rm2 = (const float*)d_in[17];
  const float* rv2 = (const float*)d_in[18];

  const int Fin = 128, H1 = 4, C1 = 64, D1 = 256, Emb = 128;
  const int N    = in_sizes[0] / Fin;     // 20000
  const int E    = in_sizes[1] / 2;       // 640000
  const int Etot = E + N;                 // +self loops
  const int T = 256;
  const float NEG_INF = -3.0e38f;

  // ---- workspace layout (floats) ----
  float* ws   = (float*)d_ws;
  float* hbuf = ws;                              // N*D1  (h1; later h2 + agg2)
  float* agg1 = hbuf + (size_t)N * D1;           // N*D1  (agg1 -> y after post)
  float* ex   = agg1 + (size_t)N * D1;           // Etot*H1 (edge logits/exp)
  float* sc   = ex   + (size_t)Etot * H1;        // 4*N*H1 scores/max/denom region
  float* ssrc1 = sc;
  float* sdst1 = sc + (size_t)N * H1;
  float* m1    = sc + 2 * (size_t)N * H1;
  float* den1  = sc + 3 * (size_t)N * H1;
  float* sums  = sc + 4 * (size_t)N * H1;        // 16*128
  float* cnt   = sums + NUM_GRAPHS * Emb;        // 16

  // =================== Layer 1 (heads=4, C=64, concat) ===================
  fill_kernel<<<cdiv(N * D1, T), T, 0, stream>>>(agg1, 0.f, N * D1);
  fill_kernel<<<cdiv(N * H1, T), T, 0, stream>>>(m1, NEG_INF, N * H1);
  fill_kernel<<<cdiv(N * H1, T), T, 0, stream>>>(den1, 0.f, N * H1);

  {
    int waves = cdiv(N, 16) * (D1 / 16);
    wmma_gemm_f32<<<cdiv(waves * 32, T), T, 0, stream>>>(x, W1, hbuf, N, D1, Fin);
  }
  scores_kernel<<<cdiv(N * H1, T), T, 0, stream>>>(hbuf, as1, ad1, ssrc1, sdst1, N, H1, C1);
  edge_logits_kernel<<<cdiv(Etot, T), T, 0, stream>>>(ei, E, Etot, ssrc1, sdst1, ex, m1, H1);
  edge_exp_kernel<<<cdiv(Etot, T), T, 0, stream>>>(ei, E, Etot, ex, m1, den1, H1);
  edge_msg_kernel<<<Etot, H1 * C1, 0, stream>>>(ei, E, ex, den1, hbuf, agg1, H1, C1);
  post_kernel<<<cdiv(N * D1, T), T, 0, stream>>>(agg1, b1, g1, be1, rm1, rv1, N, D1, 1);

  // =================== Layer 2 (heads=1, C=128) ===================
  float* h2    = hbuf;                    // reuse (h1 dead)
  float* agg2  = hbuf + (size_t)N * Emb;  // second half of hbuf region
  float* ssrc2 = sc;
  float* sdst2 = sc + (size_t)N;
  float* m2    = sc + 2 * (size_t)N;
  float* den2  = sc + 3 * (size_t)N;

  fill_kernel<<<cdiv(N * Emb, T), T, 0, stream>>>(agg2, 0.f, N * Emb);
  fill_kernel<<<cdiv(N, T), T, 0, stream>>>(m2, NEG_INF, N);
  fill_kernel<<<cdiv(N, T), T, 0, stream>>>(den2, 0.f, N);

  {
    int waves = cdiv(N, 16) * (Emb / 16);
    wmma_gemm_f32<<<cdiv(waves * 32, T), T, 0, stream>>>(agg1, W2, h2, N, Emb, D1);
  }
  scores_kernel<<<cdiv(N, T), T, 0, stream>>>(h2, as2, ad2, ssrc2, sdst2, N, 1, Emb);
  edge_logits_kernel<<<cdiv(Etot, T), T, 0, stream>>>(ei, E, Etot, ssrc2, sdst2, ex, m2, 1);
  edge_exp_kernel<<<cdiv(Etot, T), T, 0, stream>>>(ei, E, Etot, ex, m2, den2, 1);
  edge_msg_kernel<<<Etot, Emb, 0, stream>>>(ei, E, ex, den2, h2, agg2, 1, Emb);
  post_kernel<<<cdiv(N * Emb, T), T, 0, stream>>>(agg2, b2, g2, be2, rm2, rv2, N, Emb, 0);

  // =================== Global mean pool ===================
  fill_kernel<<<cdiv(NUM_GRAPHS * Emb, T), T, 0, stream>>>(sums, 0.f, NUM_GRAPHS * Emb);
  fill_kernel<<<1, T, 0, stream>>>(cnt, 0.f, NUM_GRAPHS);
  pool_accum_kernel<<<cdiv(N, 64), T, 0, stream>>>(agg2, bat, sums, N, Emb);
  count_kernel<<<cdiv(N, T), T, 0, stream>>>(bat, cnt, N);
  finalize_kernel<<<cdiv(NUM_GRAPHS * Emb, T), T, 0, stream>>>(sums, cnt, (float*)d_out,
                                                               NUM_GRAPHS, Emb);
}